// LIFJ_15496242004560
// MI455X (gfx1250) — compile-verified
//
#include <hip/hip_runtime.h>
#include <stdint.h>
#include <stddef.h>

// ---------------------------------------------------------------------------
// LIF feed-forward scan, T x [B*N] streaming recurrence.
//   v' = v + 0.1f*((0 - v) + i);  i' = 0.8f*i;  z = (v' > 1);  v = z?0:v';
//   i = i' + x[t]
// Memory-bound: 512 MB traffic -> ~22 us floor at 23.3 TB/s.
// Each thread owns 4 neurons; latency hidden by a depth-16 async global->LDS
// pipeline (ASYNCcnt), per-wave-private LDS slices so no barriers are needed.
// ---------------------------------------------------------------------------

typedef float v4f __attribute__((ext_vector_type(4)));
typedef int   v4i __attribute__((ext_vector_type(4)));
typedef __attribute__((address_space(1))) v4i as1_v4i;
typedef __attribute__((address_space(3))) v4i as3_v4i;

#define LIF_TPB   256
#define LIF_DEPTH 16   // keep in sync with "s_wait_asynccnt 15" below

#if defined(__HIP_DEVICE_COMPILE__) && defined(__gfx1250__) && \
    __has_builtin(__builtin_amdgcn_global_load_async_to_lds_b128)
#define LIF_ASYNC 1
#else
#define LIF_ASYNC 0
#endif

__device__ __forceinline__ void lif_step(float xk, float& v, float& cur, float& z) {
  // Exact rounding order of the reference (no FMA contraction): threshold is
  // a hard compare, so bitwise-matching arithmetic matters.
  float s  = __fadd_rn(cur, -v);             // (v_leak - v) + i, v_leak = 0
  float vd = __fadd_rn(v, __fmul_rn(0.1f, s));
  float id = __fmul_rn(cur, 0.8f);           // i * (1 - dt*tau_syn_inv)
  bool fire = vd > 1.0f;
  z   = fire ? 1.0f : 0.0f;
  v   = fire ? 0.0f : vd;                    // (1-z)*v' + z*v_reset, v_reset=0
  cur = __fadd_rn(id, xk);
}

// Pipelined kernel: REQUIRES T >= LIF_DEPTH (host guarantees this).
__global__ __launch_bounds__(LIF_TPB)
void lif_fwd_kernel(const float* __restrict__ x, float* __restrict__ out,
                    int T, int BN)
{
  const int gid = blockIdx.x * LIF_TPB + threadIdx.x; // float4 column id
  const int col = gid << 2;
  if (col >= BN) return;

  float v[4]   = {0.0f, 0.0f, 0.0f, 0.0f};
  float cur[4] = {0.0f, 0.0f, 0.0f, 0.0f};

#if LIF_ASYNC
  // Per-block staging: 16 slots x 256 threads x 16B = 64 KB LDS.
  // Each wave's 32 lanes occupy a contiguous 512B span per slot and only ever
  // read what they themselves requested -> no cross-wave sync needed.
  __shared__ v4f tile[LIF_DEPTH][LIF_TPB];
  const int tid = threadIdx.x;

  // Prologue: fill the pipeline (unconditional: T >= LIF_DEPTH).
#pragma unroll
  for (int s = 0; s < LIF_DEPTH; ++s) {
    const float* gp = x + (size_t)s * (size_t)BN + col;
    __builtin_amdgcn_global_load_async_to_lds_b128(
        (as1_v4i*)((void*)gp),
        (as3_v4i*)((void*)&tile[s][tid]), /*offset=*/0, /*cpol=*/0);
  }

  int t = 0;
  const int Tmain = T - LIF_DEPTH;
  for (; t < Tmain; ++t) {
    // Async loads complete in order: <=15 outstanding => slot for t has landed.
    asm volatile("s_wait_asynccnt 15" ::: "memory");
    v4f xv = tile[t & (LIF_DEPTH - 1)][tid];
    // Drain the DS read before the async engine overwrites this slot
    // (async LDS writes are unordered vs this wave's DS reads).
    asm volatile("s_wait_dscnt 0" ::: "memory");
    const float* gp = x + (size_t)(t + LIF_DEPTH) * (size_t)BN + col;
    __builtin_amdgcn_global_load_async_to_lds_b128(
        (as1_v4i*)((void*)gp),
        (as3_v4i*)((void*)&tile[t & (LIF_DEPTH - 1)][tid]), 0, 0);

    v4f zv;
#pragma unroll
    for (int k = 0; k < 4; ++k) {
      float z; lif_step(xv[k], v[k], cur[k], z); zv[k] = z;
    }
    __builtin_nontemporal_store(zv, (v4f*)(out + (size_t)t * (size_t)BN + col));
  }

  // Epilogue: everything already issued; wait for all remaining arrivals once.
  asm volatile("s_wait_asynccnt 0" ::: "memory");
  for (; t < T; ++t) {
    v4f xv = tile[t & (LIF_DEPTH - 1)][tid];
    v4f zv;
#pragma unroll
    for (int k = 0; k < 4; ++k) {
      float z; lif_step(xv[k], v[k], cur[k], z); zv[k] = z;
    }
    __builtin_nontemporal_store(zv, (v4f*)(out + (size_t)t * (size_t)BN + col));
  }
#else
  // Fallback: register-pipelined streaming loads.
#pragma unroll 4
  for (int t = 0; t < T; ++t) {
    v4f xv = __builtin_nontemporal_load(
        (const v4f*)(x + (size_t)t * (size_t)BN + col));
    v4f zv;
#pragma unroll
    for (int k = 0; k < 4; ++k) {
      float z; lif_step(xv[k], v[k], cur[k], z); zv[k] = z;
    }
    __builtin_nontemporal_store(zv, (v4f*)(out + (size_t)t * (size_t)BN + col));
  }
#endif
}

// Degenerate-case kernel (T < LIF_DEPTH): plain streaming, no LDS pipeline.
__global__ __launch_bounds__(LIF_TPB)
void lif_fwd_simple_kernel(const float* __restrict__ x, float* __restrict__ out,
                           int T, int BN)
{
  const int gid = blockIdx.x * LIF_TPB + threadIdx.x;
  const int col = gid << 2;
  if (col >= BN) return;

  float v[4]   = {0.0f, 0.0f, 0.0f, 0.0f};
  float cur[4] = {0.0f, 0.0f, 0.0f, 0.0f};

  for (int t = 0; t < T; ++t) {
    v4f xv = __builtin_nontemporal_load(
        (const v4f*)(x + (size_t)t * (size_t)BN + col));
    v4f zv;
#pragma unroll
    for (int k = 0; k < 4; ++k) {
      float z; lif_step(xv[k], v[k], cur[k], z); zv[k] = z;
    }
    __builtin_nontemporal_store(zv, (v4f*)(out + (size_t)t * (size_t)BN + col));
  }
}

extern "C" void kernel_launch(void* const* d_in, const int* in_sizes, int n_in,
                              void* d_out, int out_size, void* d_ws, size_t ws_size,
                              hipStream_t stream) {
  (void)n_in; (void)out_size; (void)d_ws; (void)ws_size;
  const float* x = (const float*)d_in[0];
  float* out = (float*)d_out;

  const int BN = 64 * 4096;            // B * N from the reference
  const int T  = in_sizes[0] / BN;     // 256
  const int cols4 = BN / 4;            // one float4 per thread

  dim3 block(LIF_TPB);
  dim3 grid((cols4 + LIF_TPB - 1) / LIF_TPB);
  if (T >= LIF_DEPTH) {
    lif_fwd_kernel<<<grid, block, 0, stream>>>(x, out, T, BN);
  } else {
    lif_fwd_simple_kernel<<<grid, block, 0, stream>>>(x, out, T, BN);
  }
}